// NeuralDictionaryV7_19430432047763
// MI455X (gfx1250) — compile-verified
//
#include <hip/hip_runtime.h>

typedef __attribute__((ext_vector_type(2))) float v2f;
typedef __attribute__((ext_vector_type(8))) float v8f;

#define DIM 256
#define CHUNKS 64  // DIM / 4

__device__ __forceinline__ unsigned long long shfl_xor_u64(unsigned long long v, int mask) {
    unsigned lo = (unsigned)(v & 0xffffffffull);
    unsigned hi = (unsigned)(v >> 32);
    lo = __shfl_xor(lo, mask, 32);
    hi = __shfl_xor(hi, mask, 32);
    return ((unsigned long long)hi << 32) | (unsigned long long)lo;
}

__global__ void nn_init_ws(unsigned long long* ws) {
    ws[0] = ~0ull;
}

// Distance + argmin. Each wave owns 16-row tiles of `keys`; the global load
// pattern lands directly in the WMMA B-layout (no cross-lane shuffles):
//   lane n    (<16) : keys[row0+n][4c+0 .. 4c+1]   (B VGPR0/1, K=0,1)
//   lane n+16       : keys[row0+n][4c+2 .. 4c+3]   (B VGPR0/1, K=2,3)
// A = query chunk replicated across M (same split per lane half), from LDS.
// dist[r] = ||keys[r]||^2 - 2 * keys[r].q  (||q||^2 dropped: constant in argmin)
__global__ __launch_bounds__(256) void nn_dist_kernel(const float* __restrict__ query,
                                                      const float* __restrict__ keys,
                                                      unsigned long long* __restrict__ best_ws,
                                                      int n) {
    __shared__ float qs[DIM];
    for (int i = threadIdx.x; i < DIM; i += blockDim.x) qs[i] = query[i];
    __syncthreads();

    const int lane          = threadIdx.x & 31;
    const int l15           = lane & 15;
    const int loff          = (lane < 16) ? 0 : 2;
    const int waveInBlock   = threadIdx.x >> 5;
    const int wavesPerBlock = blockDim.x >> 5;
    const int gw            = blockIdx.x * wavesPerBlock + waveInBlock;
    const int numWaves      = gridDim.x * wavesPerBlock;
    const int numTiles      = (n + 15) >> 4;

    // qp[2c] = { q[4c+loff], q[4c+loff+1] }  (LDS broadcast per lane half)
    const float2* qp = (const float2*)(qs + loff);

    float    bestDist = __builtin_inff();
    unsigned bestIdx  = 0u;

    for (int tile = gw; tile < numTiles; tile += numWaves) {
        const int row = tile * 16 + l15;
        const int rc  = (row < n) ? row : (n - 1);  // clamp loads; excluded from argmin
        const float2* kp = (const float2*)(keys + (size_t)rc * DIM + loff);

        v8f   c0 = {};  // even chunks
        v8f   c1 = {};  // odd chunks
        float ss = 0.0f;

#pragma unroll 8
        for (int c = 0; c < CHUNKS; c += 2) {
            const float2 k0 = kp[2 * c];
            const float2 k1 = kp[2 * c + 2];
            const float2 q0 = qp[2 * c];
            const float2 q1 = qp[2 * c + 2];

            ss = fmaf(k0.x, k0.x, ss);
            ss = fmaf(k0.y, k0.y, ss);
            ss = fmaf(k1.x, k1.x, ss);
            ss = fmaf(k1.y, k1.y, ss);

            v2f a0, b0, a1, b1;
            a0.x = q0.x; a0.y = q0.y;
            b0.x = k0.x; b0.y = k0.y;
            a1.x = q1.x; a1.y = q1.y;
            b1.x = k1.x; b1.y = k1.y;

            c0 = __builtin_amdgcn_wmma_f32_16x16x4_f32(false, a0, false, b0, (short)0, c0, false, false);
            c1 = __builtin_amdgcn_wmma_f32_16x16x4_f32(false, a1, false, b1, (short)0, c1, false, false);
        }

        // ||k||^2 is split across the lane pair (n, n+16): recombine once per tile.
        const float ssp = ss + __shfl_xor(ss, 16, 32);
        // C[0][n] == C[8][n] (A rows replicated): element 0 holds this row's dot
        // in both lane halves.
        const float dot  = c0[0] + c1[0];
        const float dist = fmaf(-2.0f, dot, ssp);
        if (row < n && dist < bestDist) { bestDist = dist; bestIdx = (unsigned)row; }
    }

    // Order-preserving u64 key: monotone float->uint map in high bits, row index
    // low (ties -> lowest index, matching argmin). Lane pairs carry duplicates;
    // the min reduction absorbs them.
    unsigned du  = __float_as_uint(bestDist);
    unsigned key = (du & 0x80000000u) ? ~du : (du | 0x80000000u);
    unsigned long long packed = ((unsigned long long)key << 32) | (unsigned long long)bestIdx;

    for (int m = 16; m > 0; m >>= 1) {
        unsigned long long o = shfl_xor_u64(packed, m);
        packed = (o < packed) ? o : packed;
    }
    if (lane == 0) atomicMin(best_ws, packed);
}

// out[0] = values[idx] . query
__global__ __launch_bounds__(256) void nn_finalize_kernel(const unsigned long long* __restrict__ ws,
                                                          const float* __restrict__ values,
                                                          const float* __restrict__ query,
                                                          float* __restrict__ out) {
    __shared__ float red[DIM];
    const unsigned idx = (unsigned)(ws[0] & 0xffffffffull);
    const int t = threadIdx.x;
    red[t] = values[(size_t)idx * DIM + t] * query[t];
    __syncthreads();
    for (int s = 128; s > 0; s >>= 1) {
        if (t < s) red[t] += red[t + s];
        __syncthreads();
    }
    if (t == 0) out[0] = red[0];
}

extern "C" void kernel_launch(void* const* d_in, const int* in_sizes, int n_in,
                              void* d_out, int out_size, void* d_ws, size_t ws_size,
                              hipStream_t stream) {
    const float* query  = (const float*)d_in[0];
    const float* keys   = (const float*)d_in[1];
    const float* values = (const float*)d_in[2];
    float* out = (float*)d_out;
    unsigned long long* ws = (unsigned long long*)d_ws;
    const int n = in_sizes[1] / DIM;

    nn_init_ws<<<dim3(1), dim3(1), 0, stream>>>(ws);
    nn_dist_kernel<<<dim3(512), dim3(256), 0, stream>>>(query, keys, ws, n);
    nn_finalize_kernel<<<dim3(1), dim3(256), 0, stream>>>(ws, values, query, out);
}